// RIOS_H_59184649338948
// MI455X (gfx1250) — compile-verified
//
#include <hip/hip_runtime.h>
#include <math.h>

// ---------------------------------------------------------------------------
// GRU-D style imputation scan for MI455X (gfx1250, wave32, WMMA).
// All three GEMMs (Whh0, Whh1 recurrences + pointer net W_p) run on
// V_WMMA_F32_16X16X4_F32 (full f32 matrix pipe, matches f32 reference).
// ---------------------------------------------------------------------------

typedef float v2f __attribute__((ext_vector_type(2)));
typedef float v8f __attribute__((ext_vector_type(8)));

constexpr int NB = 128;      // batch
constexpr int NT = 256;      // sequence length
constexpr int NH = 512;      // hidden
constexpr int NWIN = 5;      // attention half-window
constexpr int NBH = NB * NH;

// workspace layout (floats)
constexpr int OFF_H0   = 0;               // 2 buffers (double buffered)
constexpr int OFF_H1   = 2 * NBH;         // 2 buffers
constexpr int OFF_C0   = 4 * NBH;
constexpr int OFF_C1   = 5 * NBH;
constexpr int OFF_CTX  = 6 * NBH;
constexpr int OFF_Y    = 7 * NBH;         // pointer-net pre-activation y = h1 @ W_p.T
constexpr int OFF_XC   = 8 * NBH;         // NB
constexpr int OFF_SQ   = 8 * NBH + NB;    // NB
constexpr int OFF_LOSS = 8 * NBH + 2*NB;  // 1
constexpr int OFF_ENC  = 8 * NBH + 2*NB + 1;   // NB*NT*NH
constexpr int ZERO_N   = OFF_ENC;         // region to zero each call
// total floats needed: OFF_ENC + NB*NT*NH  (~69 MB)

__device__ __forceinline__ float sigf(float x) { return 1.f / (1.f + expf(-x)); }

// ---------------------------------------------------------------------------
// Core: one wave computes a 16(batch) x 16(hidden) tile of all four gates:
//   gates[b][g*H + n] += sum_k h[b,k] * Whh[g*H+n, k]   (i.e. h @ Whh.T)
// A (16x4 f32): lanes 0-15 hold M=lane, VGPR0/1 + halfwave select K
// B (4x16 f32): lanes hold N, VGPR0/1 + halfwave select K  -> both operands
// are 2 consecutive f32 along K in row-major storage.
// ---------------------------------------------------------------------------
__device__ __forceinline__ void wmma_gates4(const float* __restrict__ hcur,
                                            const float* __restrict__ Whh,
                                            int b0, int n0, int lane,
                                            v8f acc[4]) {
  const int half = lane >> 4;
  const int lrow = lane & 15;
  const v2f* __restrict__ h2 = (const v2f*)hcur;
  const v2f* __restrict__ w2 = (const v2f*)Whh;
  const int aBase = (((b0 + lrow) * NH) >> 1) + half;
  int bBase[4];
#pragma unroll
  for (int g = 0; g < 4; ++g)
    bBase[g] = (((g * NH + n0 + lrow) * NH) >> 1) + half;

#pragma unroll 4
  for (int kk = 0; kk < (NH / 2); kk += 2) {   // kk advances 2 v2f = 4 K
    v2f a = h2[aBase + kk];
#pragma unroll
    for (int g = 0; g < 4; ++g) {
      v2f bf = w2[bBase[g] + kk];
      acc[g] = __builtin_amdgcn_wmma_f32_16x16x4_f32(
          false, a, false, bf, (short)0, acc[g], false, false);
    }
  }
}

// ---------------------------------------------------------------------------
// LSTM layer 1: gates GEMM + cell + gamma decay + enc write
// grid (8, 4) x 256 threads; wave -> 16x16 tile.
// ---------------------------------------------------------------------------
__global__ __launch_bounds__(256) void lstm1_kernel(
    const float* __restrict__ values, const float* __restrict__ deltas,
    const float* __restrict__ Wih1, const float* __restrict__ Whh1,
    const float* __restrict__ bih1, const float* __restrict__ bhh1,
    const float* __restrict__ Wd, const float* __restrict__ bd,
    float* __restrict__ ws, int t) {
  const int wave = threadIdx.x >> 5;
  const int lane = threadIdx.x & 31;
  const int b0 = blockIdx.x * 16;
  const int n0 = blockIdx.y * 128 + wave * 16;

  const float* hcur  = ws + OFF_H1 + (t & 1) * NBH;
  float*       hnext = ws + OFF_H1 + ((t + 1) & 1) * NBH;
  float*       c1    = ws + OFF_C1;
  float*       enc   = ws + OFF_ENC;

  v8f acc[4] = {};
  wmma_gates4(hcur, Whh1, b0, n0, lane, acc);

  const int half = lane >> 4, lrow = lane & 15;
  const int hcol = n0 + lrow;
  const float wdv = Wd[hcol], bdv = bd[hcol];

#pragma unroll
  for (int r = 0; r < 8; ++r) {
    const int b = b0 + r + half * 8;          // C/D layout: M = r + 8*halfwave
    const float* vrow = values + (b * NT + t) * 3;
    const float xv0 = vrow[0], xv1 = vrow[1];
    float gate[4];
#pragma unroll
    for (int g = 0; g < 4; ++g) {
      const int row = g * NH + hcol;
      gate[g] = acc[g][r] + xv0 * Wih1[row * 2] + xv1 * Wih1[row * 2 + 1]
              + bih1[row] + bhh1[row];
    }
    const float iv = sigf(gate[0]);
    const float fv = sigf(gate[1]);
    const float gv = tanhf(gate[2]);
    const float ov = sigf(gate[3]);
    const int idx = b * NH + hcol;
    const float cn = fv * c1[idx] + iv * gv;
    const float hn = ov * tanhf(cn);
    const float dlt = deltas[(b * NT + t) * 3 + 2];
    const float gamma = expf(-fmaxf(0.f, dlt * wdv + bdv));
    const float hv = hn * gamma;
    c1[idx] = cn;
    hnext[idx] = hv;
    enc[(b * NT + t) * NH + hcol] = hv;
  }
}

// ---------------------------------------------------------------------------
// Pointer-net GEMM on WMMA: y = h1_new @ W_p.T   (NB x NH, K = NH)
// ---------------------------------------------------------------------------
__global__ __launch_bounds__(256) void ptr_gemm_kernel(
    const float* __restrict__ W_p, float* __restrict__ ws, int t) {
  const int wave = threadIdx.x >> 5;
  const int lane = threadIdx.x & 31;
  const int b0 = blockIdx.x * 16;
  const int n0 = blockIdx.y * 128 + wave * 16;
  const float* h = ws + OFF_H1 + ((t + 1) & 1) * NBH;   // new h1
  float* y = ws + OFF_Y;

  const int half = lane >> 4, lrow = lane & 15;
  const v2f* __restrict__ h2 = (const v2f*)h;
  const v2f* __restrict__ w2 = (const v2f*)W_p;
  const int aBase = (((b0 + lrow) * NH) >> 1) + half;
  const int bBase = (((n0 + lrow) * NH) >> 1) + half;

  v8f acc = {};
#pragma unroll 8
  for (int kk = 0; kk < (NH / 2); kk += 2) {
    v2f a = h2[aBase + kk];
    v2f bf = w2[bBase + kk];
    acc = __builtin_amdgcn_wmma_f32_16x16x4_f32(
        false, a, false, bf, (short)0, acc, false, false);
  }
  const int hcol = n0 + lrow;
#pragma unroll
  for (int r = 0; r < 8; ++r) {
    const int b = b0 + r + half * 8;
    y[b * NH + hcol] = acc[r];
  }
}

// ---------------------------------------------------------------------------
// LSTM layer 0: gates GEMM + cell (c_in = ctx for t>0)
// ---------------------------------------------------------------------------
__global__ __launch_bounds__(256) void lstm0_kernel(
    const float* __restrict__ masks,
    const float* __restrict__ Wih0, const float* __restrict__ Whh0,
    const float* __restrict__ bih0, const float* __restrict__ bhh0,
    float* __restrict__ ws, int t) {
  const int wave = threadIdx.x >> 5;
  const int lane = threadIdx.x & 31;
  const int b0 = blockIdx.x * 16;
  const int n0 = blockIdx.y * 128 + wave * 16;

  const float* hcur  = ws + OFF_H0 + (t & 1) * NBH;
  float*       hnext = ws + OFF_H0 + ((t + 1) & 1) * NBH;
  float*       c0    = ws + OFF_C0;
  const float* ctx   = ws + OFF_CTX;
  const float* xc    = ws + OFF_XC;

  v8f acc[4] = {};
  wmma_gates4(hcur, Whh0, b0, n0, lane, acc);

  const int half = lane >> 4, lrow = lane & 15;
  const int hcol = n0 + lrow;

#pragma unroll
  for (int r = 0; r < 8; ++r) {
    const int b = b0 + r + half * 8;
    const float x0 = xc[b];
    const float x1 = masks[b * NT + t];
    float gate[4];
#pragma unroll
    for (int g = 0; g < 4; ++g) {
      const int row = g * NH + hcol;
      gate[g] = acc[g][r] + x0 * Wih0[row * 2] + x1 * Wih0[row * 2 + 1]
              + bih0[row] + bhh0[row];
    }
    const float iv = sigf(gate[0]);
    const float fv = sigf(gate[1]);
    const float gv = tanhf(gate[2]);
    const float ov = sigf(gate[3]);
    const int idx = b * NH + hcol;
    const float cin = (t > 0) ? ctx[idx] : c0[idx];
    const float cn = fv * cin + iv * gv;
    const float hn = ov * tanhf(cn);
    c0[idx] = cn;
    hnext[idx] = hn;
  }
}

__device__ __forceinline__ float block_reduce256(float v, float* red) {
  const int tid = threadIdx.x;
  red[tid] = v;
  __syncthreads();
  for (int s = 128; s > 0; s >>= 1) {
    if (tid < s) red[tid] += red[tid + s];
    __syncthreads();
  }
  const float r = red[0];
  __syncthreads();
  return r;
}

// ---------------------------------------------------------------------------
// Pointer head + windowed attention + imputation. One block per batch row.
// (heavy GEMM already done by ptr_gemm_kernel; here only O(H) work)
// ---------------------------------------------------------------------------
__global__ __launch_bounds__(256) void attn_kernel(
    const float* __restrict__ values, const float* __restrict__ masks,
    const float* __restrict__ Whr, const float* __restrict__ bhr,
    const float* __restrict__ v_p, const float* __restrict__ w_attn,
    float* __restrict__ ws, float* __restrict__ out_imp, int t) {
  __shared__ float red[256];
  __shared__ float sc[2 * NWIN];

  const int b = blockIdx.x;
  const int tid = threadIdx.x;
  const float* h0c = ws + OFF_H0 + (t & 1) * NBH + b * NH;  // pre-update h0
  const float* y   = ws + OFF_Y + b * NH;                   // h1_new @ W_p.T
  const float* enc = ws + OFF_ENC;
  float* ctx = ws + OFF_CTX;

  // pointer logit: sum_j tanh(y[j]) * v_p[j]
  float part = 0.f;
  for (int j = tid; j < NH; j += 256) part += tanhf(y[j]) * v_p[j];
  const float sv = block_reduce256(part, red);

  // regression head x_h = h0 . Whr + bhr
  float xpart = 0.f;
  for (int k = tid; k < NH; k += 256) xpart += h0c[k] * Whr[k];
  const float xh = block_reduce256(xpart, red) + bhr[0];

  // window bounds (round-to-nearest-even like jnp.round)
  const float p = (float)NT * sigf(sv);
  const float pr = fminf(fmaxf(rintf(p), 0.f), (float)(NT - 1));
  const int lo = max((int)pr - NWIN, 0);
  const int hi = min(min((int)pr + NWIN, NT) - 1, t);
  const int nw = (hi >= lo) ? (hi - lo + 1) : 0;

  if (nw > 0) {
    for (int w = 0; w < nw; ++w) {
      const float* e = enc + (b * NT + lo + w) * NH;
      float s = 0.f;
      for (int k = tid; k < NH; k += 256) s += e[k] * w_attn[k];
      s = block_reduce256(s, red);
      if (tid == 0) sc[w] = s;
    }
    __syncthreads();
    float mx = -1e30f;
    for (int w = 0; w < nw; ++w) mx = fmaxf(mx, sc[w]);
    float aw[2 * NWIN];
    float den = 0.f;
    for (int w = 0; w < nw; ++w) { aw[w] = expf(sc[w] - mx); den += aw[w]; }
    const float inv = 1.f / den;
    for (int h = tid; h < NH; h += 256) {
      float a = 0.f;
      for (int w = 0; w < nw; ++w) a += aw[w] * enc[(b * NT + lo + w) * NH + h];
      ctx[b * NH + h] = a * inv;
    }
  } else {
    // empty window: reference softmax degenerates to uniform over valid [0..t]
    const float inv = 1.f / (float)(t + 1);
    for (int h = tid; h < NH; h += 256) {
      float a = 0.f;
      for (int tp = 0; tp <= t; ++tp) a += enc[(b * NT + tp) * NH + h];
      ctx[b * NH + h] = a * inv;
    }
  }

  if (tid == 0) {
    const float xy = values[(b * NT + t) * 3 + 2];
    const float m = masks[b * NT + t];
    const float xcv = m * xy + (1.f - m) * xh;
    ws[OFF_XC + b] = xcv;
    const float d = xy - xh;
    ws[OFF_SQ + b] = d * d * m;
    out_imp[b * NT + t] = xcv;
  }
}

__global__ __launch_bounds__(128) void loss_kernel(
    const float* __restrict__ masks, float* __restrict__ ws, int t) {
  __shared__ float r1[128], r2[128];
  const int tid = threadIdx.x;
  r1[tid] = masks[tid * NT + t];
  r2[tid] = ws[OFF_SQ + tid];
  __syncthreads();
  for (int s = 64; s > 0; s >>= 1) {
    if (tid < s) { r1[tid] += r1[tid + s]; r2[tid] += r2[tid + s]; }
    __syncthreads();
  }
  if (tid == 0) ws[OFF_LOSS] += r2[0] / (r1[0] + 1e-5f);
}

__global__ __launch_bounds__(256) void init_kernel(float* __restrict__ ws) {
  const int i = blockIdx.x * blockDim.x + threadIdx.x;
  if (i < ZERO_N) ws[i] = 0.f;
}

__global__ void finalize_kernel(const float* __restrict__ ws,
                                float* __restrict__ out) {
  out[0] = ws[OFF_LOSS];
}

extern "C" void kernel_launch(void* const* d_in, const int* in_sizes, int n_in,
                              void* d_out, int out_size, void* d_ws, size_t ws_size,
                              hipStream_t stream) {
  const float* values = (const float*)d_in[0];
  const float* masks  = (const float*)d_in[1];
  const float* deltas = (const float*)d_in[2];
  const float* Wih0 = (const float*)d_in[3];
  const float* Whh0 = (const float*)d_in[4];
  const float* bih0 = (const float*)d_in[5];
  const float* bhh0 = (const float*)d_in[6];
  const float* Wih1 = (const float*)d_in[7];
  const float* Whh1 = (const float*)d_in[8];
  const float* bih1 = (const float*)d_in[9];
  const float* bhh1 = (const float*)d_in[10];
  const float* Wd   = (const float*)d_in[11];
  const float* bd   = (const float*)d_in[12];
  const float* Whr  = (const float*)d_in[13];
  const float* bhr  = (const float*)d_in[14];
  const float* W_p  = (const float*)d_in[15];
  const float* v_p  = (const float*)d_in[16];
  const float* w_attn = (const float*)d_in[17];
  float* out = (float*)d_out;
  float* ws = (float*)d_ws;

  init_kernel<<<dim3((ZERO_N + 255) / 256), 256, 0, stream>>>(ws);

  for (int t = 0; t < NT; ++t) {
    lstm1_kernel<<<dim3(NB / 16, NH / 128), 256, 0, stream>>>(
        values, deltas, Wih1, Whh1, bih1, bhh1, Wd, bd, ws, t);
    ptr_gemm_kernel<<<dim3(NB / 16, NH / 128), 256, 0, stream>>>(W_p, ws, t);
    attn_kernel<<<dim3(NB), 256, 0, stream>>>(
        values, masks, Whr, bhr, v_p, w_attn, ws, out + 1, t);
    loss_kernel<<<1, 128, 0, stream>>>(masks, ws, t);
    lstm0_kernel<<<dim3(NB / 16, NH / 128), 256, 0, stream>>>(
        masks, Wih0, Whh0, bih0, bhh0, ws, t);
  }

  finalize_kernel<<<1, 1, 0, stream>>>(ws, out);
}